// PhysicsInformedKinematicLoss_62208306315814
// MI455X (gfx1250) — compile-verified
//
#include <hip/hip_runtime.h>

// ---------------- problem constants ----------------
#define KDT     0.01f
#define KALPHA  0.01f
#define KETA    0.001f

static constexpr int BATCH = 512;
static constexpr int SEQ   = 16384;
static constexpr int BLOCK = 512;                  // 16 waves (wave32)
static constexpr int WAVES = BLOCK / 32;           // 16
static constexpr int SUB   = 256;                  // elements per wave per chunk (one 16x16 tile)
static constexpr int CHUNK = WAVES * SUB;          // 4096
static constexpr int NCH   = SEQ / CHUNK;          // 4
static constexpr int PGROW = 20;                   // 16 data floats + 4 pad (keeps 16B alignment, kills bank conflicts)
static constexpr int PGTSZ = (CHUNK / 16) * PGROW; // 5120 floats per p_gt buffer

typedef __attribute__((ext_vector_type(2))) float v2f;
typedef __attribute__((ext_vector_type(8))) float v8f;

// async-copy pointer types, matching the builtin's signature:
//   arg0: AS1 (global, printed as __device__) pointer to 16B int vector
//   arg1: AS3 (LDS, printed as __shared__)   pointer to 16B int vector
typedef __attribute__((__vector_size__(16))) int v4i;
typedef v4i __attribute__((address_space(1)))* gv4i_p;
typedef v4i __attribute__((address_space(3)))* lv4i_p;

#if defined(__HIP_DEVICE_COMPILE__) && __has_builtin(__builtin_amdgcn_global_load_async_to_lds_b128) && __has_builtin(__builtin_amdgcn_s_wait_asynccnt)
#define KIN_ASYNC 1
#else
#define KIN_ASYNC 0
#endif

// Stage one 4096-float chunk of p_gt into LDS, 16->20 padded rows, cooperatively.
// Each wave issues exactly 2 async instructions (tracked by its own ASYNCcnt).
__device__ __forceinline__ void stage_pgt_chunk(const float* __restrict__ src,
                                                float* __restrict__ dst, int tid) {
  #pragma unroll
  for (int r = 0; r < 2; ++r) {
    const int g  = tid + r * BLOCK;              // 4-float group id: 0..1023
    const int fi = g * 4;                        // source float index within chunk
    const int di = (fi >> 4) * PGROW + (fi & 15);
#if KIN_ASYNC
    __builtin_amdgcn_global_load_async_to_lds_b128(
        (gv4i_p)(src + fi),
        (lv4i_p)(dst + di),
        0 /*imm offset*/, 0 /*cpol*/);
#else
    *(float4*)(dst + di) = *(const float4*)(src + fi);
#endif
  }
}

__global__ __launch_bounds__(BLOCK) void kin_loss_pass1(
    const float*  __restrict__ a_dirty,
    const float4* __restrict__ states,
    const float*  __restrict__ p_gt,
    const float*  __restrict__ v_0,
    const float*  __restrict__ p_0,
    float*        __restrict__ partials)
{
  __shared__ __align__(16) float s_pgt[2][PGTSZ]; // 40 KB: double-buffered p_gt
  __shared__ float s_ac[WAVES][272];              // a_clean tiles, 17-stride (conflict-free)
  __shared__ float s_wt[WAVES][2];                // per-wave (A, W) aggregates
  __shared__ float s_red[2][WAVES];

  const int b    = blockIdx.x;
  const int tid  = threadIdx.x;
  const int lane = tid & 31;
  const int w    = tid >> 5;
  const int hi16 = lane >> 4;      // 0: lanes 0-15, 1: lanes 16-31
  const int nn   = lane & 15;
  const int rowbase = b * SEQ;

  // Constant triangular A-operands in the documented 16x4 f32 A layout:
  //   VGPR0 holds K = {0,2} (low/high lane half), VGPR1 holds K = {1,3}.
  // L[m][k] = (k<=m)          -> running sum (S1)
  // T[m][k] = (k<=m)?(m-k+1)  -> running sum of running sum (S2)
  v2f AL[4], AT[4];
  #pragma unroll
  for (int kb = 0; kb < 4; ++kb) {
    const int m  = nn;
    const int k0 = kb * 4 + hi16 * 2;
    AL[kb].x = (k0     <= m) ? 1.0f : 0.0f;
    AL[kb].y = (k0 + 1 <= m) ? 1.0f : 0.0f;
    AT[kb].x = (k0     <= m) ? (float)(m - k0 + 1) : 0.0f;
    AT[kb].y = (k0 + 1 <= m) ? (float)(m - k0)     : 0.0f;
  }

  const float v0 = v_0[b];
  const float p0 = p_0[b];

  float G1 = 0.f, G2 = 0.f;      // row-level carries (identical across threads)
  float lp = 0.f, dmg = 0.f;     // per-thread loss partials

  stage_pgt_chunk(p_gt + rowbase, s_pgt[0], tid);

  for (int it = 0; it < NCH; ++it) {
    const int cbase = rowbase + it * CHUNK;
    if (it + 1 < NCH)
      stage_pgt_chunk(p_gt + cbase + CHUNK, s_pgt[(it + 1) & 1], tid);

    // ---- compute a_clean for this wave's 256 elements, store transposed into LDS ----
    float* ac = s_ac[w];
    #pragma unroll
    for (int q = 0; q < 8; ++q) {
      const int es = q * 32 + lane;             // position within this wave's tile
      const int g  = cbase + w * SUB + es;      // coalesced global index
      if (it + 1 < NCH) {                       // stream prefetch of next chunk
        __builtin_prefetch(a_dirty + g + CHUNK, 0, 1);
        __builtin_prefetch(states  + g + CHUNK, 0, 1);
      }
      const float  a  = a_dirty[g];
      const float4 st = states[g];              // (D, DTmp, sigma, creep)
      const float ks  = (1.f - st.x) * (1.f + KALPHA * st.y) * (1.f - KETA * st.z);
      const float acl = (a - st.w) / (ks + 1e-6f);
      dmg += fmaxf(-st.x, 0.f);
      ac[(es & 15) * 17 + (es >> 4)] = acl;     // X[k][n] at k*17+n
    }

#if KIN_ASYNC
    if (it + 1 < NCH) __builtin_amdgcn_s_wait_asynccnt(2);  // current chunk landed, next in flight
    else              __builtin_amdgcn_s_wait_asynccnt(0);
#endif
    __syncthreads();

    // ---- 16 segment scans at once: D1 = L*X, D2 = T*X via chained 16x16x4 f32 WMMA ----
    v8f d1 = {0.f,0.f,0.f,0.f,0.f,0.f,0.f,0.f};
    v8f d2 = {0.f,0.f,0.f,0.f,0.f,0.f,0.f,0.f};
    #pragma unroll
    for (int kb = 0; kb < 4; ++kb) {
      const int k0 = kb * 4 + hi16 * 2;         // B layout mirrors A: VGPR0 K={0,2}, VGPR1 K={1,3}
      v2f bb;
      bb.x = ac[(k0    ) * 17 + nn];
      bb.y = ac[(k0 + 1) * 17 + nn];
      d1 = __builtin_amdgcn_wmma_f32_16x16x4_f32(false, AL[kb], false, bb, (short)0, d1, false, false);
      d2 = __builtin_amdgcn_wmma_f32_16x16x4_f32(false, AT[kb], false, bb, (short)0, d2, false, false);
    }

    // ---- carries across the 16 segments of this wave (seg totals live in lanes 16+n, v7) ----
    float c1 = 0.f, c2 = 0.f, C1 = 0.f, C2 = 0.f;
    #pragma unroll
    for (int u = 0; u < 16; ++u) {
      const float au = __shfl(d1[7], 16 + u, 32);
      const float wu = __shfl(d2[7], 16 + u, 32);
      if (u == nn) { C1 = c1; C2 = c2; }
      c2 += 16.f * c1 + wu;                     // S2' = S2 + L*S1 + W, L = 16
      c1 += au;
    }

    if (lane == 0) { s_wt[w][0] = c1; s_wt[w][1] = c2; }
    __syncthreads();

    // ---- carries across the 16 waves of this chunk (L = 256) ----
    float WC1 = 0.f, WC2 = 0.f, t1 = 0.f, t2 = 0.f;
    #pragma unroll
    for (int u = 0; u < WAVES; ++u) {
      const float au = s_wt[u][0];
      const float wu = s_wt[u][1];
      if (u == w) { WC1 = t1; WC2 = t2; }
      t2 += 256.f * t1 + wu;
      t1 += au;
    }

    // ---- per-element loss: p_pred = p0 + (t+1)*DT*v0 + DT^2 * S2[t] ----
    const float* pgbuf = s_pgt[it & 1];
    #pragma unroll
    for (int v = 0; v < 8; ++v) {
      const int m  = v + hi16 * 8;              // C/D layout: VGPRv holds rows v / v+8
      const int es = nn * 16 + m;
      const int e  = w * SUB + es;
      const float S2 = G2 + (float)(e  + 1) * G1
                     + WC2 + (float)(es + 1) * WC1
                     + C2  + (float)(m  + 1) * C1 + d2[v];
      const int t = it * CHUNK + e;
      const float p_pred = p0 + (float)(t + 1) * KDT * v0 + (KDT * KDT) * S2;
      const float diff = p_pred - pgbuf[(e >> 4) * PGROW + (e & 15)];
      lp += diff * diff;
    }

    G2 += (float)CHUNK * G1 + t2;               // advance row carries (L = 4096)
    G1 += t1;
    __syncthreads();                            // protect s_wt & s_pgt[it&1] before reuse
  }

  // ---- deterministic block reduction -> per-row partial (loss_pos, loss_dmg) ----
  #pragma unroll
  for (int o = 16; o > 0; o >>= 1) {
    lp  += __shfl_down(lp,  o, 32);
    dmg += __shfl_down(dmg, o, 32);
  }
  if (lane == 0) { s_red[0][w] = lp; s_red[1][w] = dmg; }
  __syncthreads();
  if (tid == 0) {
    float L = 0.f, D = 0.f;
    for (int u = 0; u < WAVES; ++u) { L += s_red[0][u]; D += s_red[1][u]; }
    partials[2 * b]     = L;
    partials[2 * b + 1] = D;
  }
}

__global__ __launch_bounds__(256) void kin_loss_pass2(const float* __restrict__ partials,
                                                      float* __restrict__ out)
{
  __shared__ float sl[256], sd[256];
  const int t = threadIdx.x;
  float lp = 0.f, dm = 0.f;
  for (int i = t; i < BATCH; i += 256) { lp += partials[2 * i]; dm += partials[2 * i + 1]; }
  sl[t] = lp; sd[t] = dm;
  __syncthreads();
  for (int s = 128; s > 0; s >>= 1) {
    if (t < s) { sl[t] += sl[t + s]; sd[t] += sd[t + s]; }
    __syncthreads();
  }
  if (t == 0) {
    const float invN = 1.0f / ((float)BATCH * (float)SEQ);
    out[0] = sl[0] * invN + 0.1f * (sd[0] * invN);
  }
}

extern "C" void kernel_launch(void* const* d_in, const int* in_sizes, int n_in,
                              void* d_out, int out_size, void* d_ws, size_t ws_size,
                              hipStream_t stream) {
  const float*  a  = (const float*)d_in[0];
  const float4* st = (const float4*)d_in[1];
  const float*  pg = (const float*)d_in[2];
  const float*  v0 = (const float*)d_in[3];
  const float*  p0 = (const float*)d_in[4];
  float* partials  = (float*)d_ws;              // 512 * 2 floats = 4 KB
  float* out       = (float*)d_out;

  kin_loss_pass1<<<BATCH, BLOCK, 0, stream>>>(a, st, pg, v0, p0, partials);
  kin_loss_pass2<<<1, 256, 0, stream>>>(partials, out);
}